// DM_32530082300094
// MI455X (gfx1250) — compile-verified
//
#include <hip/hip_runtime.h>
#include <hip/hip_bf16.h>

// ---------------------------------------------------------------------------
// CDNA5 (gfx1250) fp32 WMMA: V_WMMA_F32_16X16X4_F32
//   A frag: v2f per lane (lane<16: M=lane, K=k0,k0+1 ; lane>=16: K=k0+2,k0+3)
//   B frag: v2f per lane (mirrored, N=lane&15)
//   C/D:    v8f per lane (m = m0+v+8*(lane>>4), n = n0+(lane&15))
// All WMMA sites run with full 32-lane waves; bounds handled by INDEX CLAMPS
// (garbage rows/cols never stored), never by predicating loads -> no exec-mask
// branching in the hot loops.
// ---------------------------------------------------------------------------
typedef __attribute__((ext_vector_type(2))) float v2f;
typedef __attribute__((ext_vector_type(8))) float v8f;

__device__ __forceinline__ v8f wmma4(v2f a, v2f b, v8f c) {
  return __builtin_amdgcn_wmma_f32_16x16x4_f32(false, a, false, b, (short)0, c,
                                               false, false);
}
__device__ __forceinline__ float sigf(float x) { return 1.0f / (1.0f + __expf(-x)); }

// ---------------------------------------------------------------------------
// Generic WMMA GEMM.  C[M,Nx2tiles] = act( A op B + bias ), batch = blockIdx.z.
//   TA==0: A(m,k)=A[m*lda+k]    TA==1: A(m,k)=A[k*lda+m]
//   TB==1: B(k,n)=B[n*ldb+k]    TB==0: B(k,n)=B[k*ldb+n]
//   ACT: 0 none, 1 tanh, 2 sigmoid.   REQUIRES K % 4 == 0, lda/ldb even.
// One wave per 16x32 C block (2 N-tiles share the A fragment).
// ---------------------------------------------------------------------------
template <int TA, int TB, int ACT>
__global__ void gemm_k(const float* __restrict__ Ag, const float* __restrict__ Bg,
                       float* __restrict__ Cg, const float* __restrict__ bias,
                       int M, int N, int K, int lda, int ldb, int ldc,
                       long long sA, long long sB, long long sC) {
  const float* A = Ag + (long long)blockIdx.z * sA;
  const float* B = Bg + (long long)blockIdx.z * sB;
  float* C = Cg + (long long)blockIdx.z * sC;
  int m0 = blockIdx.y << 4, n0 = blockIdx.x << 5;
  int lane = threadIdx.x & 31;
  int l15 = lane & 15, hi = lane >> 4;
  int ma = m0 + l15; if (ma >= M) ma = M - 1;
  int nb0 = n0 + l15, nb1 = n0 + 16 + l15;
  int nc0 = nb0 < N ? nb0 : N - 1;
  int nc1 = nb1 < N ? nb1 : N - 1;
  const int koff = hi << 1;
  v8f acc0 = {}, acc1 = {};
  const float* arow = TA ? (A + ma) : (A + (long long)ma * lda);
  const float* br0 = TB ? (B + (long long)nc0 * ldb) : (B + nc0);
  const float* br1 = TB ? (B + (long long)nc1 * ldb) : (B + nc1);
#pragma unroll 4
  for (int k0 = 0; k0 < K; k0 += 4) {
    int ka = k0 + koff;
    v2f av, bv0, bv1;
    if (TA) {
      av.x = arow[(long long)ka * lda];
      av.y = arow[(long long)(ka + 1) * lda];
    } else {
      av = *(const v2f*)(arow + ka);
    }
    if (TB) {
      bv0 = *(const v2f*)(br0 + ka);
      bv1 = *(const v2f*)(br1 + ka);
    } else {
      bv0.x = br0[(long long)ka * ldb];
      bv0.y = br0[(long long)(ka + 1) * ldb];
      bv1.x = br1[(long long)ka * ldb];
      bv1.y = br1[(long long)(ka + 1) * ldb];
    }
    acc0 = wmma4(av, bv0, acc0);
    acc1 = wmma4(av, bv1, acc1);
  }
  float bs0 = bias ? bias[nc0] : 0.f;
  float bs1 = bias ? bias[nc1] : 0.f;
  if (nb0 < N) {
#pragma unroll
    for (int v = 0; v < 8; ++v) {
      int m = m0 + v + (hi << 3);
      if (m < M) {
        float x = acc0[v] + bs0;
        if (ACT == 1) x = tanhf(x);
        if (ACT == 2) x = sigf(x);
        C[(long long)m * ldc + nb0] = x;
      }
    }
  }
  if (nb1 < N) {
#pragma unroll
    for (int v = 0; v < 8; ++v) {
      int m = m0 + v + (hi << 3);
      if (m < M) {
        float x = acc1[v] + bs1;
        if (ACT == 1) x = tanhf(x);
        if (ACT == 2) x = sigf(x);
        C[(long long)m * ldc + nb1] = x;
      }
    }
  }
}

// ---------------------------------------------------------------------------
// BiLSTM scan.  grid.x = 2 (fwd/bwd blocks).  h in LDS (zero-padded to LS_HP
// cols -> K-tail reads zeros, no guards), c in registers, gates via WMMA.
// Bn is 16 or 48 and exactly covers the M-tiles -> no batch-index guards.
// ---------------------------------------------------------------------------
#define LS_B 48
#define LS_HP 152
__global__ __launch_bounds__(256) void lstm_scan(
    const float* __restrict__ xgf, const float* __restrict__ xgb,
    const float* __restrict__ Whhf, const float* __restrict__ Whhb,
    float* __restrict__ out, int Bn, int T, int h, int ostride) {
  __shared__ float hl[LS_B * LS_HP];
  int dir = blockIdx.x;
  const float* xg = dir ? xgb : xgf;
  const float* Whh = dir ? Whhb : Whhf;
  int ooff = dir * h;
  int tid = threadIdx.x, lane = tid & 31, wave = tid >> 5, nw = blockDim.x >> 5;
  int l15 = lane & 15, hi = lane >> 4;
  int J = (h + 15) >> 4, MT = (Bn + 15) >> 4, units = J * MT;
  int g4 = 4 * h;
  int koff = hi << 1;
  int kmain = h & ~3;
  for (int i = tid; i < Bn * LS_HP; i += blockDim.x) hl[i] = 0.f;
  __syncthreads();
  float creg[4][8], hnew[4][8];
#pragma unroll
  for (int u = 0; u < 4; ++u)
#pragma unroll
    for (int v = 0; v < 8; ++v) { creg[u][v] = 0.f; hnew[u][v] = 0.f; }

  for (int step = 0; step < T; ++step) {
    int t = dir ? (T - 1 - step) : step;
#pragma unroll
    for (int u = 0; u < 4; ++u) {
      int unit = wave + u * nw;
      if (unit < units) {
        int j0 = (unit % J) << 4, m0 = (unit / J) << 4;
        int ma = m0 + l15;
        int jn = j0 + l15;
        int jnc = jn < h ? jn : h - 1;
        const float* w0 = Whh + (long long)(0 * h + jnc) * h;
        const float* w1 = Whh + (long long)(1 * h + jnc) * h;
        const float* w2 = Whh + (long long)(2 * h + jnc) * h;
        const float* w3 = Whh + (long long)(3 * h + jnc) * h;
        const float* hrow = &hl[ma * LS_HP];
        v8f z = {};
        v8f acc[4]; acc[0] = z; acc[1] = z; acc[2] = z; acc[3] = z;
#pragma unroll 2
        for (int k0 = 0; k0 < kmain; k0 += 4) {
          int ka = k0 + koff;
          v2f av = *(const v2f*)(hrow + ka);
          acc[0] = wmma4(av, *(const v2f*)(w0 + ka), acc[0]);
          acc[1] = wmma4(av, *(const v2f*)(w1 + ka), acc[1]);
          acc[2] = wmma4(av, *(const v2f*)(w2 + ka), acc[2]);
          acc[3] = wmma4(av, *(const v2f*)(w3 + ka), acc[3]);
        }
        if (h & 3) {  // tail: av reads zero-pad, so B value is don't-care
          int ka = kmain + koff;
          int kb = ka < h ? ka : 0;
          v2f av = *(const v2f*)(hrow + ka);
          acc[0] = wmma4(av, *(const v2f*)(w0 + kb), acc[0]);
          acc[1] = wmma4(av, *(const v2f*)(w1 + kb), acc[1]);
          acc[2] = wmma4(av, *(const v2f*)(w2 + kb), acc[2]);
          acc[3] = wmma4(av, *(const v2f*)(w3 + kb), acc[3]);
        }
#pragma unroll
        for (int v = 0; v < 8; ++v) {
          int m = m0 + v + (hi << 3);
          const float* xr = xg + ((long long)m * T + t) * g4;
          float gi = acc[0][v] + xr[jnc];
          float gf = acc[1][v] + xr[h + jnc];
          float gg = acc[2][v] + xr[2 * h + jnc];
          float go = acc[3][v] + xr[3 * h + jnc];
          float c = sigf(gf) * creg[u][v] + sigf(gi) * tanhf(gg);
          creg[u][v] = c;
          hnew[u][v] = sigf(go) * tanhf(c);
        }
      }
    }
    __syncthreads();
#pragma unroll
    for (int u = 0; u < 4; ++u) {
      int unit = wave + u * nw;
      if (unit < units) {
        int j0 = (unit % J) << 4, m0 = (unit / J) << 4;
        int jn = j0 + l15;
        if (jn < h) {
#pragma unroll
          for (int v = 0; v < 8; ++v) {
            int m = m0 + v + (hi << 3);
            hl[m * LS_HP + jn] = hnew[u][v];
            out[((long long)m * T + t) * ostride + ooff + jn] = hnew[u][v];
          }
        }
      }
    }
    __syncthreads();
  }
}

// ---------------------------------------------------------------------------
// BiGRU scan (gate order r,z,n).  xg includes bih; bhh added here.  h==128.
// ---------------------------------------------------------------------------
__global__ __launch_bounds__(256) void gru_scan(
    const float* __restrict__ xgf, const float* __restrict__ xgb,
    const float* __restrict__ Whhf, const float* __restrict__ Whhb,
    const float* __restrict__ bhhf, const float* __restrict__ bhhb,
    float* __restrict__ out, int Bn, int T, int h, int ostride, int obase) {
  __shared__ float hl[LS_B * LS_HP];
  int dir = blockIdx.x;
  const float* xg = dir ? xgb : xgf;
  const float* Whh = dir ? Whhb : Whhf;
  const float* bhh = dir ? bhhb : bhhf;
  int ooff = obase + dir * h;
  int tid = threadIdx.x, lane = tid & 31, wave = tid >> 5, nw = blockDim.x >> 5;
  int l15 = lane & 15, hi = lane >> 4;
  int J = (h + 15) >> 4, MT = (Bn + 15) >> 4, units = J * MT;
  int g3 = 3 * h;
  int koff = hi << 1;
  for (int i = tid; i < Bn * LS_HP; i += blockDim.x) hl[i] = 0.f;
  __syncthreads();
  float hnew[2][8];
#pragma unroll
  for (int u = 0; u < 2; ++u)
#pragma unroll
    for (int v = 0; v < 8; ++v) hnew[u][v] = 0.f;

  for (int step = 0; step < T; ++step) {
    int t = dir ? (T - 1 - step) : step;
#pragma unroll
    for (int u = 0; u < 2; ++u) {
      int unit = wave + u * nw;
      if (unit < units) {
        int j0 = (unit % J) << 4, m0 = (unit / J) << 4;
        int ma = m0 + l15;
        int jn = j0 + l15;
        int jnc = jn < h ? jn : h - 1;
        const float* w0 = Whh + (long long)(0 * h + jnc) * h;
        const float* w1 = Whh + (long long)(1 * h + jnc) * h;
        const float* w2 = Whh + (long long)(2 * h + jnc) * h;
        const float* hrow = &hl[ma * LS_HP];
        v8f z8 = {};
        v8f acc[3]; acc[0] = z8; acc[1] = z8; acc[2] = z8;
#pragma unroll 2
        for (int k0 = 0; k0 < h; k0 += 4) {  // h % 4 == 0 for all GRUs
          int ka = k0 + koff;
          v2f av = *(const v2f*)(hrow + ka);
          acc[0] = wmma4(av, *(const v2f*)(w0 + ka), acc[0]);
          acc[1] = wmma4(av, *(const v2f*)(w1 + ka), acc[1]);
          acc[2] = wmma4(av, *(const v2f*)(w2 + ka), acc[2]);
        }
        float b0 = bhh[jnc], b1 = bhh[h + jnc], b2 = bhh[2 * h + jnc];
#pragma unroll
        for (int v = 0; v < 8; ++v) {
          int m = m0 + v + (hi << 3);
          const float* xr = xg + ((long long)m * T + t) * g3;
          float r = sigf(xr[jnc] + acc[0][v] + b0);
          float zz = sigf(xr[h + jnc] + acc[1][v] + b1);
          float n = tanhf(xr[2 * h + jnc] + r * (acc[2][v] + b2));
          float hp = hl[m * LS_HP + jnc];
          hnew[u][v] = (1.f - zz) * n + zz * hp;
        }
      }
    }
    __syncthreads();
#pragma unroll
    for (int u = 0; u < 2; ++u) {
      int unit = wave + u * nw;
      if (unit < units) {
        int j0 = (unit % J) << 4, m0 = (unit / J) << 4;
        int jn = j0 + l15;
        if (jn < h) {
#pragma unroll
          for (int v = 0; v < 8; ++v) {
            int m = m0 + v + (hi << 3);
            hl[m * LS_HP + jn] = hnew[u][v];
            out[((long long)m * T + t) * ostride + ooff + jn] = hnew[u][v];
          }
        }
      }
    }
    __syncthreads();
  }
}

// ---------------------------------------------------------------------------
// Fused bilinear attention: sd[b,d,q] = sum_h vd[h]*tanh(sum_k Wd[h,k]*D[b,d,k]*Q[b,q,k])
// grid (DL, B), 256 threads.  A(m,k)=Q[q,k]*D[d,k] built on the fly; q index
// clamped (garbage rows dropped in epilogue).
// ---------------------------------------------------------------------------
__global__ __launch_bounds__(256) void fused_md_attn(
    const float* __restrict__ Q, const float* __restrict__ D,
    const float* __restrict__ Wd, const float* __restrict__ vd,
    float* __restrict__ sd, int QLn, int K2, int Hh) {
  int d = blockIdx.x, b = blockIdx.y;
  __shared__ float Dd[256];
  __shared__ float accs[64];
  int tid = threadIdx.x;
  for (int k = tid; k < K2; k += blockDim.x)
    Dd[k] = D[((long long)b * gridDim.x + d) * K2 + k];
  for (int i = tid; i < 64; i += blockDim.x) accs[i] = 0.f;
  __syncthreads();
  int lane = tid & 31, wave = tid >> 5, nw = blockDim.x >> 5;
  int l15 = lane & 15, hi = lane >> 4;
  int koff = hi << 1;
  int MTt = (QLn + 15) >> 4;
  int NT = Hh >> 4;
  for (int tile = wave; tile < MTt * NT; tile += nw) {
    int m0 = (tile / NT) << 4, n0 = (tile % NT) << 4;
    int mq = m0 + l15; if (mq >= QLn) mq = QLn - 1;
    int nh = n0 + l15;
    v8f acc = {};
    const float* qrow = Q + ((long long)b * QLn + mq) * K2;
    const float* wrow = Wd + (long long)nh * K2;
#pragma unroll 2
    for (int k0 = 0; k0 < K2; k0 += 4) {
      int ka = k0 + koff;
      v2f qa = *(const v2f*)(qrow + ka);
      v2f da = *(const v2f*)(&Dd[ka]);
      v2f av; av.x = qa.x * da.x; av.y = qa.y * da.y;
      v2f bv = *(const v2f*)(wrow + ka);
      acc = wmma4(av, bv, acc);
    }
    float vn = vd[nh];
#pragma unroll
    for (int v = 0; v < 8; ++v) {
      float p = vn * tanhf(acc[v]);
      for (int o = 8; o >= 1; o >>= 1) p += __shfl_xor(p, o, 16);
      if (l15 == 0) {
        int m = m0 + v + (hi << 3);
        if (m < QLn) atomicAdd(&accs[m], p);
      }
    }
  }
  __syncthreads();
  for (int qj = tid; qj < QLn; qj += blockDim.x)
    sd[((long long)b * gridDim.x + d) * QLn + qj] = accs[qj];
}

// --------------------------- small utility kernels -------------------------
__global__ void gather_rows(const float* __restrict__ emb, const int* __restrict__ idx,
                            float* __restrict__ out, int E) {
  int r = blockIdx.x;
  int t = idx[r];
  const float* s = emb + (long long)t * E;
  float* o = out + (long long)r * E;
  for (int e = threadIdx.x; e < E; e += blockDim.x) o[e] = s[e];
}

__global__ void row_softmax(const float* __restrict__ in, float* __restrict__ out, int C) {
  int r = blockIdx.x, lane = threadIdx.x;
  const float* row = in + (long long)r * C;
  float mx = -1e30f;
  for (int c = lane; c < C; c += 32) mx = fmaxf(mx, row[c]);
  for (int o = 16; o >= 1; o >>= 1) mx = fmaxf(mx, __shfl_xor(mx, o, 32));
  float s = 0.f;
  for (int c = lane; c < C; c += 32) s += __expf(row[c] - mx);
  for (int o = 16; o >= 1; o >>= 1) s += __shfl_xor(s, o, 32);
  float inv = 1.f / s;
  float* orow = out + (long long)r * C;
  for (int c = lane; c < C; c += 32) orow[c] = __expf(row[c] - mx) * inv;
}

// softmax over d of L[b,d,q], written transposed: out[b,q,d]
__global__ void col_softmax_T(const float* __restrict__ L, float* __restrict__ out,
                              int D, int Q) {
  int q = blockIdx.x, b = blockIdx.y, lane = threadIdx.x;
  const float* base = L + (long long)b * D * Q + q;
  float mx = -1e30f;
  for (int d = lane; d < D; d += 32) mx = fmaxf(mx, base[(long long)d * Q]);
  for (int o = 16; o >= 1; o >>= 1) mx = fmaxf(mx, __shfl_xor(mx, o, 32));
  float s = 0.f;
  for (int d = lane; d < D; d += 32) s += __expf(base[(long long)d * Q] - mx);
  for (int o = 16; o >= 1; o >>= 1) s += __shfl_xor(s, o, 32);
  float inv = 1.f / s;
  float* o2 = out + ((long long)b * Q + q) * D;
  for (int d = lane; d < D; d += 32) o2[d] = __expf(base[(long long)d * Q] - mx) * inv;
}

__global__ void rowdot(const float* __restrict__ X, const float* __restrict__ v,
                       float* __restrict__ out, int C) {
  int r = blockIdx.x, lane = threadIdx.x;
  const float* x = X + (long long)r * C;
  float s = 0.f;
  for (int c = lane; c < C; c += 32) s += x[c] * v[c];
  for (int o = 16; o >= 1; o >>= 1) s += __shfl_xor(s, o, 32);
  if (lane == 0) out[r] = s;
}

// out[z,i,j] = sum_h v[h]*tanh(sx*X[z,i,h] + Y[z,j,h]); grid (I, Z)
__global__ void pairwise_tanh_dot(const float* __restrict__ X, const float* __restrict__ Y,
                                  const float* __restrict__ v, float* __restrict__ out,
                                  int I, int J, int H, float sx) {
  int i = blockIdx.x, z = blockIdx.y;
  __shared__ float xs[128];
  for (int h0 = threadIdx.x; h0 < H; h0 += blockDim.x)
    xs[h0] = sx * X[((long long)z * I + i) * H + h0];
  __syncthreads();
  for (int j = threadIdx.x; j < J; j += blockDim.x) {
    const float* y = Y + ((long long)z * J + j) * H;
    float s = 0.f;
    for (int h0 = 0; h0 < H; ++h0) s += v[h0] * tanhf(xs[h0] + y[h0]);
    out[((long long)z * I + i) * J + j] = s;
  }
}

// out[n,e] = sum_a P[n,a]*X[n,a,e]
__global__ void attn_pool(const float* __restrict__ P, const float* __restrict__ X,
                          float* __restrict__ out, int A, int E) {
  int n = blockIdx.x;
  for (int e = threadIdx.x; e < E; e += blockDim.x) {
    float s = 0.f;
    for (int a = 0; a < A; ++a)
      s += P[(long long)n * A + a] * X[((long long)n * A + a) * E + e];
    out[(long long)n * E + e] = s;
  }
}

__global__ void copy_cols(const float* __restrict__ src, float* __restrict__ dst,
                          int C, int lds_, int ldd, int off) {
  int r = blockIdx.x;
  const float* s = src + (long long)r * lds_;
  float* d = dst + (long long)r * ldd + off;
  for (int c = threadIdx.x; c < C; c += blockDim.x) d[c] = s[c];
}

__global__ void mul_ew(const float* __restrict__ a, const float* __restrict__ b,
                       float* __restrict__ c, long long n) {
  long long i = (long long)blockIdx.x * blockDim.x + threadIdx.x;
  if (i < n) c[i] = a[i] * b[i];
}

// slog[b*3+a] = dot(Aemb[b*3+a,:], Enc[b,:])
__global__ void score_logits(const float* __restrict__ Aemb, const float* __restrict__ Enc,
                             float* __restrict__ out, int E) {
  int a = blockIdx.x, b = blockIdx.y, lane = threadIdx.x;
  const float* x = Aemb + ((long long)b * 3 + a) * E;
  const float* y = Enc + (long long)b * E;
  float s = 0.f;
  for (int e = lane; e < E; e += 32) s += x[e] * y[e];
  for (int o = 16; o >= 1; o >>= 1) s += __shfl_xor(s, o, 32);
  if (lane == 0) out[b * 3 + a] = s;
}

// ---------------------------------------------------------------------------
// Host-side GEMM dispatcher
// ---------------------------------------------------------------------------
static void gemm(hipStream_t st, int TA, int TB, int ACT, const float* A,
                 const float* B, float* C, const float* bias, int M, int N, int K,
                 int lda, int ldb, int ldc, int batch = 1, long long sA = 0,
                 long long sB = 0, long long sC = 0) {
  dim3 g((N + 31) / 32, (M + 15) / 16, batch), blk(32);
#define GEMM_L(ta, tb, ac) \
  gemm_k<ta, tb, ac><<<g, blk, 0, st>>>(A, B, C, bias, M, N, K, lda, ldb, ldc, sA, sB, sC)
  if (TA == 0 && TB == 1 && ACT == 0) GEMM_L(0, 1, 0);
  else if (TA == 0 && TB == 1 && ACT == 1) GEMM_L(0, 1, 1);
  else if (TA == 0 && TB == 1 && ACT == 2) GEMM_L(0, 1, 2);
  else if (TA == 0 && TB == 0 && ACT == 0) GEMM_L(0, 0, 0);
  else if (TA == 1 && TB == 0 && ACT == 0) GEMM_L(1, 0, 0);
#undef GEMM_L
}

// ---------------------------------------------------------------------------
// kernel_launch: full forward pass
// ---------------------------------------------------------------------------
extern "C" void kernel_launch(void* const* d_in, const int* in_sizes, int n_in,
                              void* d_out, int out_size, void* d_ws, size_t ws_size,
                              hipStream_t stream) {
  (void)in_sizes; (void)n_in; (void)out_size; (void)ws_size;
  const int Bb = 16, QL = 60, DL = 400, AL = 8, E = 300;
  const int H2 = 256;  // 2*H
  const int NQ = Bb * QL, ND = Bb * DL, NA = Bb * 3;

  const int* query = (const int*)d_in[0];
  const int* passage = (const int*)d_in[1];
  const int* answer = (const int*)d_in[2];
  auto P = [&](int i) -> const float* { return (const float*)d_in[i]; };
  // param indices (dict insertion order, depth first):
  // 3 emb | 4-7 a_enc | 8-11 q_enc | 12-15 d_enc | 16-19 U_lstm | 20 a_att
  // 21 W_Q 22 b_Q 23 Wc1 24 Wc2 25 vc 26 Wb 27 Wd 28 vd 29 Wm 30 vm
  // 31 Wgc 32-39 gru_htc | 40 Wgb 41-48 gru_htb | 49 Wgd 50-57 gru_htd
  // 58 Wgm 59-66 gru_htm | 67 W_agg 68 v_agg | 69-76 gru_agg | 77-84 out_gru
  // 85 Wq 86 vq 87 Wp1 88 Wp2 89 vp 90 pred_W

  char* base = (char*)d_ws;
  size_t cur = 0;
  auto alloc = [&](size_t nfloat) -> float* {
    float* r = (float*)(base + cur);
    cur += ((nfloat * sizeof(float)) + 255) & ~(size_t)255;
    return r;
  };
  float* q_emb = alloc((size_t)NQ * E);
  float* d_emb = alloc((size_t)ND * E);
  float* a_emb = alloc((size_t)NA * AL * E);
  float* a_xgf = alloc((size_t)NA * AL * 600);
  float* a_xgb = alloc((size_t)NA * AL * 600);
  float* a_encO = alloc((size_t)NA * AL * E);
  float* a_scv = alloc((size_t)NA * AL);
  float* a_embed = alloc((size_t)NA * E);
  float* q_xgf = alloc((size_t)NQ * 512);
  float* q_xgb = alloc((size_t)NQ * 512);
  float* Qh = alloc((size_t)NQ * H2);
  float* Qm = alloc((size_t)NQ * H2);
  float* d_xgf = alloc((size_t)ND * 512);
  float* d_xgb = alloc((size_t)ND * 512);
  float* Dm = alloc((size_t)ND * H2);
  float* Lm = alloc((size_t)ND * QL);
  float* ADt = alloc((size_t)NQ * DL);
  float* CQ = alloc((size_t)NQ * H2);
  float* QCQ = alloc((size_t)NQ * 512);
  float* CD = alloc((size_t)ND * 512);
  float* DCD = alloc((size_t)ND * 1024);
  float* s1c = alloc((size_t)NQ * 128);
  float* s2c = alloc((size_t)ND * 128);
  float* scS = alloc((size_t)ND * QL);
  float* sbS = alloc((size_t)ND * QL);
  float* sdS = alloc((size_t)ND * QL);
  float* smS = alloc((size_t)ND * QL);
  float* QWb = alloc((size_t)NQ * H2);
  float* qmB = alloc((size_t)NQ * 128);
  float* dmB = alloc((size_t)ND * 128);
  float* qt = alloc((size_t)ND * H2);
  float* xb = alloc((size_t)ND * 512);
  float* gb = alloc((size_t)ND * 512);
  float* gx = alloc((size_t)ND * 512);
  float* bxgf = alloc((size_t)ND * 384);
  float* bxgb = alloc((size_t)ND * 384);
  float* agg = alloc((size_t)ND * 1024);
  float* aggW = alloc((size_t)ND * 4 * 128);
  float* sl = alloc((size_t)ND * 4);
  float* rep = alloc((size_t)ND * H2);
  float* Um = alloc((size_t)ND * H2);
  float* agg_rep = alloc((size_t)ND * H2);
  float* Qq = alloc((size_t)NQ * 128);
  float* sq = alloc((size_t)NQ);
  float* rq = alloc((size_t)Bb * H2);
  float* Up1 = alloc((size_t)ND * 128);
  float* rqW = alloc((size_t)Bb * 128);
  float* sp = alloc((size_t)ND);
  float* rp = alloc((size_t)Bb * H2);
  float* encv = alloc((size_t)Bb * E);
  float* slog = alloc((size_t)Bb * 3);
  float* out = (float*)d_out;

  // 1. embedding gathers
  gather_rows<<<NQ, 256, 0, stream>>>(P(3), query, q_emb, E);
  gather_rows<<<ND, 256, 0, stream>>>(P(3), passage, d_emb, E);
  gather_rows<<<NA * AL, 256, 0, stream>>>(P(3), answer, a_emb, E);

  // 2. answer encoder (BiLSTM, h=150)
  gemm(stream, 0, 1, 0, a_emb, P(4), a_xgf, nullptr, NA * AL, 600, E, E, E, 600);
  gemm(stream, 0, 1, 0, a_emb, P(6), a_xgb, nullptr, NA * AL, 600, E, E, E, 600);
  lstm_scan<<<2, 256, 0, stream>>>(a_xgf, a_xgb, P(5), P(7), a_encO, NA, AL, 150, 300);
  rowdot<<<NA * AL, 32, 0, stream>>>(a_encO, P(20), a_scv, E);
  row_softmax<<<NA, 32, 0, stream>>>(a_scv, a_scv, AL);
  attn_pool<<<NA, 256, 0, stream>>>(a_scv, a_encO, a_embed, AL, E);

  // 3. query encoder + projection
  gemm(stream, 0, 1, 0, q_emb, P(8), q_xgf, nullptr, NQ, 512, E, E, E, 512);
  gemm(stream, 0, 1, 0, q_emb, P(10), q_xgb, nullptr, NQ, 512, E, E, E, 512);
  lstm_scan<<<2, 256, 0, stream>>>(q_xgf, q_xgb, P(9), P(11), Qh, Bb, QL, 128, H2);
  gemm(stream, 0, 1, 1, Qh, P(21), Qm, P(22), NQ, H2, H2, H2, H2, H2);

  // 4. document encoder
  gemm(stream, 0, 1, 0, d_emb, P(12), d_xgf, nullptr, ND, 512, E, E, E, 512);
  gemm(stream, 0, 1, 0, d_emb, P(14), d_xgb, nullptr, ND, 512, E, E, E, 512);
  lstm_scan<<<2, 256, 0, stream>>>(d_xgf, d_xgb, P(13), P(15), Dm, Bb, DL, 128, H2);

  // 5. coattention
  gemm(stream, 0, 1, 0, Dm, Qm, Lm, nullptr, DL, QL, H2, H2, H2, QL, Bb,
       (long long)DL * H2, (long long)QL * H2, (long long)DL * QL);
  col_softmax_T<<<dim3(QL, Bb), 32, 0, stream>>>(Lm, ADt, DL, QL);
  row_softmax<<<ND, 32, 0, stream>>>(Lm, Lm, QL);  // AQ in place
  gemm(stream, 1, 0, 0, Lm, Dm, CQ, nullptr, QL, H2, DL, QL, H2, H2, Bb,
       (long long)DL * QL, (long long)DL * H2, (long long)QL * H2);
  copy_cols<<<NQ, 256, 0, stream>>>(Qm, QCQ, H2, H2, 512, 0);
  copy_cols<<<NQ, 256, 0, stream>>>(CQ, QCQ, H2, H2, 512, H2);
  gemm(stream, 1, 0, 0, ADt, QCQ, CD, nullptr, DL, 512, QL, DL, 512, 512, Bb,
       (long long)QL * DL, (long long)QL * 512, (long long)DL * 512);
  copy_cols<<<ND, 256, 0, stream>>>(Dm, DCD, H2, H2, 1024, 0);
  copy_cols<<<ND, 256, 0, stream>>>(Dm, DCD, H2, H2, 1024, H2);
  copy_cols<<<ND, 256, 0, stream>>>(CD, DCD, 512, 512, 1024, 512);

  // 6. four attention score tensors
  gemm(stream, 0, 1, 0, Qm, P(23), s1c, nullptr, NQ, 128, H2, H2, H2, 128);
  gemm(stream, 0, 1, 0, Dm, P(24), s2c, nullptr, ND, 128, H2, H2, H2, 128);
  pairwise_tanh_dot<<<dim3(DL, Bb), 64, 0, stream>>>(s2c, s1c, P(25), scS, DL, QL, 128, 1.f);
  gemm(stream, 0, 1, 0, Qm, P(26), QWb, nullptr, NQ, H2, H2, H2, H2, H2);
  gemm(stream, 0, 1, 0, Dm, QWb, sbS, nullptr, DL, QL, H2, H2, H2, QL, Bb,
       (long long)DL * H2, (long long)QL * H2, (long long)DL * QL);
  fused_md_attn<<<dim3(DL, Bb), 256, 0, stream>>>(Qm, Dm, P(27), P(28), sdS, QL, H2, 128);
  gemm(stream, 0, 1, 0, Qm, P(29), qmB, nullptr, NQ, 128, H2, H2, H2, 128);
  gemm(stream, 0, 1, 0, Dm, P(29), dmB, nullptr, ND, 128, H2, H2, H2, 128);
  pairwise_tanh_dot<<<dim3(DL, Bb), 64, 0, stream>>>(dmB, qmB, P(30), smS, DL, QL, 128, -1.f);
  row_softmax<<<ND, 32, 0, stream>>>(scS, scS, QL);
  row_softmax<<<ND, 32, 0, stream>>>(sbS, sbS, QL);
  row_softmax<<<ND, 32, 0, stream>>>(sdS, sdS, QL);
  row_softmax<<<ND, 32, 0, stream>>>(smS, smS, QL);

  // 7. four gated BiGRU branches -> agg (concat via column offsets)
  const float* Ps[4] = {scS, sbS, sdS, smS};
  const int wgIdx[4] = {31, 40, 49, 58};
  const int grIdx[4] = {32, 41, 50, 59};
  for (int br = 0; br < 4; ++br) {
    gemm(stream, 0, 0, 0, Ps[br], Qm, qt, nullptr, DL, H2, QL, QL, H2, H2, Bb,
         (long long)DL * QL, (long long)QL * H2, (long long)DL * H2);
    copy_cols<<<ND, 256, 0, stream>>>(qt, xb, H2, H2, 512, 0);
    copy_cols<<<ND, 256, 0, stream>>>(Dm, xb, H2, H2, 512, H2);
    gemm(stream, 0, 1, 2, xb, P(wgIdx[br]), gb, nullptr, ND, 512, 512, 512, 512, 512);
    long long nel = (long long)ND * 512;
    mul_ew<<<(unsigned)((nel + 255) / 256), 256, 0, stream>>>(gb, xb, gx, nel);
    int g0 = grIdx[br];
    gemm(stream, 0, 1, 0, gx, P(g0 + 0), bxgf, P(g0 + 2), ND, 384, 512, 512, 512, 384);
    gemm(stream, 0, 1, 0, gx, P(g0 + 4), bxgb, P(g0 + 6), ND, 384, 512, 512, 512, 384);
    gru_scan<<<2, 256, 0, stream>>>(bxgf, bxgb, P(g0 + 1), P(g0 + 5), P(g0 + 3),
                                    P(g0 + 7), agg, Bb, DL, 128, 1024, br * H2);
  }

  // 8. aggregation attention over 4 branches
  gemm(stream, 0, 1, 1, agg, P(67), aggW, nullptr, ND * 4, 128, H2, H2, H2, 128);
  rowdot<<<ND * 4, 32, 0, stream>>>(aggW, P(68), sl, 128);
  row_softmax<<<ND, 32, 0, stream>>>(sl, sl, 4);
  attn_pool<<<ND, 256, 0, stream>>>(sl, agg, rep, 4, H2);

  // 9. U = BiLSTM(D_CD)  (reuse d_xg buffers)
  gemm(stream, 0, 1, 0, DCD, P(16), d_xgf, nullptr, ND, 512, 1024, 1024, 1024, 512);
  gemm(stream, 0, 1, 0, DCD, P(18), d_xgb, nullptr, ND, 512, 1024, 1024, 1024, 512);
  lstm_scan<<<2, 256, 0, stream>>>(d_xgf, d_xgb, P(17), P(19), Um, Bb, DL, 128, H2);

  // 10. agg_rep = BiGRU(rep)
  gemm(stream, 0, 1, 0, rep, P(69), bxgf, P(71), ND, 384, H2, H2, H2, 384);
  gemm(stream, 0, 1, 0, rep, P(73), bxgb, P(75), ND, 384, H2, H2, H2, 384);
  gru_scan<<<2, 256, 0, stream>>>(bxgf, bxgb, P(70), P(74), P(72), P(76), agg_rep,
                                  Bb, DL, 128, H2, 0);

  // 11. output = BiGRU(concat(U, agg_rep)) -> d_out
  copy_cols<<<ND, 256, 0, stream>>>(Um, xb, H2, H2, 512, 0);
  copy_cols<<<ND, 256, 0, stream>>>(agg_rep, xb, H2, H2, 512, H2);
  gemm(stream, 0, 1, 0, xb, P(77), bxgf, P(79), ND, 384, 512, 512, 512, 384);
  gemm(stream, 0, 1, 0, xb, P(81), bxgb, P(83), ND, 384, 512, 512, 512, 384);
  gru_scan<<<2, 256, 0, stream>>>(bxgf, bxgb, P(78), P(82), P(80), P(84), out,
                                  Bb, DL, 128, H2, 0);

  // 12. answer scoring
  gemm(stream, 0, 1, 1, Qm, P(85), Qq, nullptr, NQ, 128, H2, H2, H2, 128);
  rowdot<<<NQ, 32, 0, stream>>>(Qq, P(86), sq, 128);
  row_softmax<<<Bb, 32, 0, stream>>>(sq, sq, QL);
  attn_pool<<<Bb, 256, 0, stream>>>(sq, Qm, rq, QL, H2);
  gemm(stream, 0, 1, 0, Um, P(87), Up1, nullptr, ND, 128, H2, H2, H2, 128);
  gemm(stream, 0, 1, 0, rq, P(88), rqW, nullptr, Bb, 128, H2, H2, H2, 128);
  pairwise_tanh_dot<<<dim3(DL, Bb), 64, 0, stream>>>(Up1, rqW, P(89), sp, DL, 1, 128, 1.f);
  row_softmax<<<Bb, 32, 0, stream>>>(sp, sp, DL);
  attn_pool<<<Bb, 256, 0, stream>>>(sp, Um, rp, DL, H2);
  gemm(stream, 0, 1, 1, rp, P(90), encv, nullptr, Bb, E, H2, H2, H2, E);
  score_logits<<<dim3(3, Bb), 32, 0, stream>>>(a_embed, encv, slog, E);
  row_softmax<<<Bb, 32, 0, stream>>>(slog, out + (size_t)Bb * DL * H2, 3);
}